// Speech_MSA_DW_11785390260918
// MI455X (gfx1250) — compile-verified
//
#include <hip/hip_runtime.h>
#include <hip/hip_bf16.h>

typedef __attribute__((ext_vector_type(2))) float v2f;
typedef __attribute__((ext_vector_type(8))) float v8f;

#define EMBED   256
#define THREE_E 768
#define HD      64
#define B_      4
#define W_      64
#define F_      4096
#define T_      4160
#define NHEADS  16          // B_ * 4 heads
#define SCALING 0.125f      // 64^-0.5

// ---------------------------------------------------------------------------
// fp32 WMMA 64x64 register-blocked GEMM: one wave computes a 64x64 block of C
// as 4x4 WMMA 16x16 tiles (acc = 128 VGPRs).  Per k-step: 4 A-fragments +
// 4 B-fragments feed 16 v_wmma_f32_16x16x4_f32 ops (4x fewer VMEM per WMMA
// than 16x16 tiling).
// A element (m,k) = A[m*rsA + k*csA]; B element (k,n) = B[k*rsB + n*csB].
// A fragment (16x4 f32): lanes 0-15 hold row m=lane, VGPR0/1 = K+0/K+1;
// lanes 16-31 hold K+2/K+3.  B fragment mirrors with column = lane&15.
// ---------------------------------------------------------------------------
__device__ __forceinline__ void wmma_f32_block64(
    const float* __restrict__ A, long rsA, long csA,
    const float* __restrict__ Bm, long rsB, long csB,
    int K, v8f acc[4][4]) {
  const int lane = threadIdx.x & 31;
  const int half = lane >> 4;
  const int l16  = lane & 15;
  const float* ap = A  + (long)l16 * rsA + (long)(2 * half) * csA;
  const float* bp = Bm + (long)(2 * half) * rsB + (long)l16 * csB;
  __builtin_prefetch(ap, 0, 1);   // global_prefetch_b8
  __builtin_prefetch(bp, 0, 1);
#pragma unroll 2
  for (int k = 0; k < K; k += 4) {
    v2f a[4], b[4];
#pragma unroll
    for (int i = 0; i < 4; ++i) {
      a[i].x = ap[(long)(i * 16) * rsA + (long)k * csA];
      a[i].y = ap[(long)(i * 16) * rsA + (long)(k + 1) * csA];
    }
#pragma unroll
    for (int j = 0; j < 4; ++j) {
      b[j].x = bp[(long)k * rsB + (long)(j * 16) * csB];
      b[j].y = bp[(long)(k + 1) * rsB + (long)(j * 16) * csB];
    }
#pragma unroll
    for (int i = 0; i < 4; ++i)
#pragma unroll
      for (int j = 0; j < 4; ++j)
        acc[i][j] = __builtin_amdgcn_wmma_f32_16x16x4_f32(
            /*neg_a=*/false, a[i], /*neg_b=*/false, b[j],
            /*c_mod=*/(short)0, acc[i][j], /*reuse_a=*/false, /*reuse_b=*/false);
  }
}

// C/D 16x16 f32 layout: VGPR r = row (r + 8*half), col = lane&15.
__device__ __forceinline__ void wmma_store_block64(float* __restrict__ C, long ldc,
                                                   v8f acc[4][4]) {
  const int lane = threadIdx.x & 31;
  const int half = lane >> 4;
  const int l16  = lane & 15;
#pragma unroll
  for (int i = 0; i < 4; ++i)
#pragma unroll
    for (int j = 0; j < 4; ++j)
#pragma unroll
      for (int r = 0; r < 8; ++r)
        C[(long)(i * 16 + r + 8 * half) * ldc + (j * 16 + l16)] = acc[i][j][r];
}

// ---------------------------------------------------------------------------
// Kernel 1: qkv = x(16640x256) @ W_qkv(256x768) + b_qkv     [WMMA f32]
// 260 x 12 blocks of 64x64, one wave each, 8 waves/block -> 390 blocks.
// ---------------------------------------------------------------------------
__global__ void __launch_bounds__(256) qkv_gemm_kernel(
    const float* __restrict__ x, const float* __restrict__ Wqkv,
    const float* __restrict__ bqkv, float* __restrict__ qkv) {
  const int wave = (blockIdx.x * blockDim.x + threadIdx.x) >> 5;
  const int NT = THREE_E / 64;            // 12 column blocks
  const int tm = wave / NT;               // 0..259
  const int tn = wave - tm * NT;
  const int l16 = threadIdx.x & 15;
  v8f acc[4][4];
#pragma unroll
  for (int j = 0; j < 4; ++j) {
    const float bv = bqkv[tn * 64 + j * 16 + l16];
#pragma unroll
    for (int i = 0; i < 4; ++i)
#pragma unroll
      for (int r = 0; r < 8; ++r) acc[i][j][r] = bv;
  }
  wmma_f32_block64(x + (long)tm * 64 * EMBED, EMBED, 1,
                   Wqkv + tn * 64, THREE_E, 1, EMBED, acc);
  wmma_store_block64(qkv + (long)tm * 64 * THREE_E + tn * 64, THREE_E, acc);
}

// ---------------------------------------------------------------------------
// Kernel 2: expa[n] (4096x64) = wm[b]^T (4096x64) @ wtok[n] (64x64)  [WMMA f32]
// wtok[n](w,dd) = x[b, w, h*64+dd]; A(f,w) = wm[b*W*F + w*F + f] (rsA=1, csA=F)
// 16 heads x 64 frame-blocks, one 64x64 block per wave -> 128 blocks.
// ---------------------------------------------------------------------------
__global__ void __launch_bounds__(256) expa_gemm_kernel(
    const float* __restrict__ x, const float* __restrict__ wm,
    float* __restrict__ expa) {
  const int wave = (blockIdx.x * blockDim.x + threadIdx.x) >> 5;
  const int ft = wave & 63;               // 0..63 frame block
  const int n  = wave >> 6;               // 0..15 head
  const int b = n >> 2, h = n & 3;
  v8f acc[4][4];
#pragma unroll
  for (int i = 0; i < 4; ++i)
#pragma unroll
    for (int j = 0; j < 4; ++j)
#pragma unroll
      for (int r = 0; r < 8; ++r) acc[i][j][r] = 0.f;
  const float* A  = wm + (long)b * W_ * F_ + ft * 64;              // (f,w)
  const float* Bm = x  + (long)b * T_ * EMBED + h * HD;            // (w,dd)
  wmma_f32_block64(A, /*rsA=*/1, /*csA=*/F_, Bm, /*rsB=*/EMBED, /*csB=*/1, W_, acc);
  wmma_store_block64(expa + ((long)n * F_ + ft * 64) * HD, HD, acc);
}

// ---------------------------------------------------------------------------
// Kernel 3: copy word tokens (t < 64) straight into the pre-projection buffer
// (heads split/recombine is identity on these rows).
// ---------------------------------------------------------------------------
__global__ void copy_wtok_kernel(const float* __restrict__ x, float* __restrict__ pre) {
  const int i = blockIdx.x * blockDim.x + threadIdx.x;   // 0..65535
  const int b = i >> 14;                                  // / (64*256)
  const int r = i & 16383;
  pre[(long)b * T_ * EMBED + r] = x[(long)b * T_ * EMBED + r];
}

// ---------------------------------------------------------------------------
// Kernel 4: local-window attention. One block = one (head, window) pair,
// 64 threads = 64 frames.  Zero-padded keys get score 0 (matches reference
// zero-pad, NOT -inf masking).  Word-token weight softmaxes over the 64
// frames of the window (LDS reduction).
// ---------------------------------------------------------------------------
__global__ void __launch_bounds__(64) attn_local_kernel(
    const float* __restrict__ qkv, const float* __restrict__ expa,
    float* __restrict__ pre) {
  const int w = blockIdx.x;               // window 0..63
  const int n = blockIdx.y;               // head 0..15
  const int b = n >> 2, h = n & 3;
  const int f = w * 64 + threadIdx.x;     // frame 0..4095

  // Load Q row (64 floats) into registers.
  const float4* qp = (const float4*)(qkv + (long)(b * T_ + W_ + f) * THREE_E + h * HD);
  float4 q4[16];
#pragma unroll
  for (int i = 0; i < 16; ++i) q4[i] = qp[i];

  const float* erow = expa + ((long)n * F_ + f) * HD;

  // score[0]: word-token key = expa row
  float s0 = 0.f;
  {
    const float4* ep = (const float4*)erow;
#pragma unroll
    for (int i = 0; i < 16; ++i) {
      const float4 e = ep[i];
      s0 += q4[i].x * e.x + q4[i].y * e.y + q4[i].z * e.z + q4[i].w * e.w;
    }
    s0 *= SCALING;
  }

  // local window scores
  float sl[15];
  for (int j = 0; j < 15; ++j) {
    const int kf = f - 7 + j;
    float a = 0.f;
    if (kf >= 0 && kf < F_) {
      const float4* kp = (const float4*)(qkv + (long)(b * T_ + W_ + kf) * THREE_E
                                         + EMBED + h * HD);
#pragma unroll
      for (int i = 0; i < 16; ++i) {
        const float4 kv = kp[i];
        a += q4[i].x * kv.x + q4[i].y * kv.y + q4[i].z * kv.z + q4[i].w * kv.w;
      }
    }
    sl[j] = a * SCALING;     // zero-padded key -> score 0
  }

  // softmax over the 15 local scores
  float m = sl[0];
#pragma unroll
  for (int j = 1; j < 15; ++j) m = fmaxf(m, sl[j]);
  float sum = 0.f;
#pragma unroll
  for (int j = 0; j < 15; ++j) { sl[j] = __expf(sl[j] - m); sum += sl[j]; }
  const float inv = 1.f / sum;
#pragma unroll
  for (int j = 0; j < 15; ++j) sl[j] *= inv;

  // word-token softmax across the 64 frames of this window
  __shared__ float smem[64];
  smem[threadIdx.x] = s0;
  __syncthreads();
  float m0 = -3.4e38f;
  for (int i = 0; i < 64; ++i) m0 = fmaxf(m0, smem[i]);
  float sum0 = 0.f;
  for (int i = 0; i < 64; ++i) sum0 += __expf(smem[i] - m0);
  const float w0 = __expf(s0 - m0) / sum0;

  // out = w0*expa + sum_j wj * V[f-7+j]
  float4 o4[16];
  {
    const float4* ep = (const float4*)erow;
#pragma unroll
    for (int i = 0; i < 16; ++i) {
      const float4 e = ep[i];
      o4[i].x = w0 * e.x; o4[i].y = w0 * e.y; o4[i].z = w0 * e.z; o4[i].w = w0 * e.w;
    }
  }
  for (int j = 0; j < 15; ++j) {
    const int kf = f - 7 + j;
    if (kf >= 0 && kf < F_) {
      const float wj = sl[j];
      const float4* vp = (const float4*)(qkv + (long)(b * T_ + W_ + kf) * THREE_E
                                         + 2 * EMBED + h * HD);
#pragma unroll
      for (int i = 0; i < 16; ++i) {
        const float4 vv = vp[i];
        o4[i].x += wj * vv.x; o4[i].y += wj * vv.y;
        o4[i].z += wj * vv.z; o4[i].w += wj * vv.w;
      }
    }
  }
  float4* op = (float4*)(pre + (long)(b * T_ + W_ + f) * EMBED + h * HD);
#pragma unroll
  for (int i = 0; i < 16; ++i) op[i] = o4[i];
}

// ---------------------------------------------------------------------------
// Kernel 5: out = pre(16640x256) @ W_out(256x256) + b_out    [WMMA f32]
// 260 x 4 blocks of 64x64 -> 130 blocks of 8 waves.
// ---------------------------------------------------------------------------
__global__ void __launch_bounds__(256) out_gemm_kernel(
    const float* __restrict__ pre, const float* __restrict__ Wout,
    const float* __restrict__ bout, float* __restrict__ out) {
  const int wave = (blockIdx.x * blockDim.x + threadIdx.x) >> 5;
  const int NT = EMBED / 64;              // 4 column blocks
  const int tm = wave / NT;               // 0..259
  const int tn = wave - tm * NT;
  const int l16 = threadIdx.x & 15;
  v8f acc[4][4];
#pragma unroll
  for (int j = 0; j < 4; ++j) {
    const float bv = bout[tn * 64 + j * 16 + l16];
#pragma unroll
    for (int i = 0; i < 4; ++i)
#pragma unroll
      for (int r = 0; r < 8; ++r) acc[i][j][r] = bv;
  }
  wmma_f32_block64(pre + (long)tm * 64 * EMBED, EMBED, 1,
                   Wout + tn * 64, EMBED, 1, EMBED, acc);
  wmma_store_block64(out + (long)tm * 64 * EMBED + tn * 64, EMBED, acc);
}

// ---------------------------------------------------------------------------
extern "C" void kernel_launch(void* const* d_in, const int* in_sizes, int n_in,
                              void* d_out, int out_size, void* d_ws, size_t ws_size,
                              hipStream_t stream) {
  (void)in_sizes; (void)n_in; (void)out_size; (void)ws_size;
  const float* x    = (const float*)d_in[0];
  const float* wm   = (const float*)d_in[1];
  const float* Wqkv = (const float*)d_in[2];
  const float* bqkv = (const float*)d_in[3];
  const float* Wout = (const float*)d_in[4];
  const float* bout = (const float*)d_in[5];
  float* out = (float*)d_out;

  // Workspace layout (floats): qkv | expa | pre  (~85 MB total)
  float* qkv  = (float*)d_ws;
  float* expa = qkv  + (long)B_ * T_ * THREE_E;     // 12,779,520
  float* pre  = expa + (long)NHEADS * F_ * HD;      // + 4,194,304

  // 1) QKV projection: 260 x 12 64x64 blocks, 8 waves/block -> 390 blocks
  qkv_gemm_kernel<<<390, 256, 0, stream>>>(x, Wqkv, bqkv, qkv);
  // 2) expa scatter-GEMM: 16 heads x 64 frame blocks -> 128 blocks
  expa_gemm_kernel<<<128, 256, 0, stream>>>(x, wm, expa);
  // 3) word-token passthrough rows of pre
  copy_wtok_kernel<<<256, 256, 0, stream>>>(x, pre);
  // 4) local attention: one block per (window, head)
  attn_local_kernel<<<dim3(64, 16), 64, 0, stream>>>(qkv, expa, pre);
  // 5) output projection: 260 x 4 64x64 blocks -> 130 blocks
  out_gemm_kernel<<<130, 256, 0, stream>>>(pre, Wout, bout, out);
}